// DynamicMaskHead_1202590843792
// MI455X (gfx1250) — compile-verified
//
#include <hip/hip_runtime.h>

typedef float v2f __attribute__((ext_vector_type(2)));
typedef float v8f __attribute__((ext_vector_type(8)));

#define CH      8
#define INCH    8
#define HH      128
#define WW      128
#define HW      (HH * WW)
#define NPARAMS 169

// x_elem: row k of the per-instance input matrix X (12x16384, zero padded from 10 rows)
__device__ __forceinline__ float x_elem(int k, int p,
                                        float sx, float sy, float inv_soi,
                                        const float* __restrict__ mask_shift,
                                        const float* __restrict__ feats) {
  float r = 0.0f;
  if (k == 0)      r = (sx - mask_shift[2 * p + 0]) * inv_soi;
  else if (k == 1) r = (sy - mask_shift[2 * p + 1]) * inv_soi;
  else if (k < 10) r = feats[(k - 2) * HW + p];
  return r;
}

// Kernel A: per-instance dynamic MLP -> logits[inst][128*128], via V_WMMA_F32_16X16X4_F32.
// grid = (n_inst, 8 pixel-chunks), block = 256 (8 waves). Each wave: 16 px / iter, 16 iters.
__global__ void __launch_bounds__(256)
mask_head_logits(const float* __restrict__ mask_feats,
                 const float* __restrict__ mask_shift,
                 const float* __restrict__ shifts,
                 const float* __restrict__ inst_params,
                 const int* __restrict__ im_inds,
                 const int* __restrict__ fpn_levels,
                 float* __restrict__ logits) {
  __shared__ float lds[8 * 128];          // 8 waves x (8 ch x 16 px) activation slab
  const int inst  = blockIdx.x;
  const int chunk = blockIdx.y;           // 2048 pixels per chunk
  const int tid   = threadIdx.x;
  const int wave  = tid >> 5;
  const int lane  = tid & 31;
  const int m     = lane & 15;            // A row / B column within tile
  const int hi    = lane >> 4;            // lane-half selects K {+0,+1} vs {+2,+3}
  float* wlds = lds + wave * 128;

  const int   lvl     = fpn_levels[inst];
  const float inv_soi = 1.0f / (64.0f * (float)(1 << lvl));
  const float sx      = shifts[inst * 2 + 0];
  const float sy      = shifts[inst * 2 + 1];
  const float* feats  = mask_feats + (size_t)im_inds[inst] * (INCH * HW);
  const float* wp     = inst_params + (size_t)inst * NPARAMS;

  // ---- hoist loop-invariant A operands (weights, zero-padded to 16 rows / K mult of 4) ----
  v2f a1[3];                               // w1: 8x10 -> 16x12
  #pragma unroll
  for (int kc = 0; kc < 3; ++kc) {
    const int k0 = kc * 4 + hi * 2;
    v2f t;
    t.x = (m < CH && k0 < 10)     ? wp[m * 10 + k0]     : 0.0f;
    t.y = (m < CH && k0 + 1 < 10) ? wp[m * 10 + k0 + 1] : 0.0f;
    a1[kc] = t;
  }
  v2f a2[2];                               // w2: 8x8 -> 16x8
  #pragma unroll
  for (int kc = 0; kc < 2; ++kc) {
    const int k0 = kc * 4 + hi * 2;
    v2f t;
    t.x = (m < CH) ? wp[80 + m * 8 + k0]     : 0.0f;
    t.y = (m < CH) ? wp[80 + m * 8 + k0 + 1] : 0.0f;
    a2[kc] = t;
  }
  float w3v[8], b1v[8], b2v[8];
  #pragma unroll
  for (int c = 0; c < 8; ++c) {
    w3v[c] = wp[144 + c];
    b1v[c] = wp[152 + c];
    b2v[c] = wp[160 + c];
  }
  const float b3 = wp[168];

  for (int it = 0; it < 16; ++it) {
    const int p = chunk * 2048 + it * 128 + wave * 16 + m;

    // ---- layer 1: D(16x16) = w1(16x12) * X(12x16), 3 WMMA K-steps ----
    v8f c1 = {};
    #pragma unroll
    for (int kc = 0; kc < 3; ++kc) {
      const int k = kc * 4 + hi * 2;
      v2f b;
      b.x = x_elem(k,     p, sx, sy, inv_soi, mask_shift, feats);
      b.y = x_elem(k + 1, p, sx, sy, inv_soi, mask_shift, feats);
      c1 = __builtin_amdgcn_wmma_f32_16x16x4_f32(false, a1[kc], false, b,
                                                 (short)0, c1, false, false);
    }
    #pragma unroll
    for (int r = 0; r < 8; ++r) c1[r] = fmaxf(c1[r] + b1v[r], 0.0f);
    if (lane < 16) {                        // D: VGPR r = channel r, lane = pixel
      #pragma unroll
      for (int r = 0; r < 8; ++r) wlds[r * 16 + lane] = c1[r];
    }

    // ---- layer 2: 2 WMMA K-steps, B from LDS (in-order per-wave LDS, no barrier) ----
    v8f c2 = {};
    #pragma unroll
    for (int kc = 0; kc < 2; ++kc) {
      const int k = kc * 4 + hi * 2;
      v2f b;
      b.x = wlds[k * 16 + m];
      b.y = wlds[(k + 1) * 16 + m];
      c2 = __builtin_amdgcn_wmma_f32_16x16x4_f32(false, a2[kc], false, b,
                                                 (short)0, c2, false, false);
    }
    #pragma unroll
    for (int r = 0; r < 8; ++r) c2[r] = fmaxf(c2[r] + b2v[r], 0.0f);
    if (lane < 16) {
      #pragma unroll
      for (int r = 0; r < 8; ++r) wlds[r * 16 + lane] = c2[r];
    }

    // ---- layer 3: 1x8 dot + bias (scalar; a WMMA would be 1/16 utilized) ----
    if (lane < 16) {
      float acc = b3;
      #pragma unroll
      for (int ch = 0; ch < 8; ++ch)
        acc = fmaf(w3v[ch], wlds[ch * 16 + lane], acc);
      logits[(size_t)inst * HW + p] = acc;
    }
  }
}

// Kernel B: factor-2 aligned_bilinear. out[y,x] = 4-tap stencil of logits with
// my=max(y-1,0), i0=my>>1, frac=(my&1)*0.5, edge-clamped. Pure store-bandwidth.
__global__ void __launch_bounds__(256)
upsample2x(const float* __restrict__ logits, float* __restrict__ out, int total) {
  const int idx = blockIdx.x * 256 + threadIdx.x;
  if (idx >= total) return;
  const int x  = idx & 255;
  const int y  = (idx >> 8) & 255;
  const int ni = idx >> 16;

  const int my = max(y - 1, 0), mx = max(x - 1, 0);
  const int i0y = my >> 1, i0x = mx >> 1;
  const int i1y = min(i0y + 1, HH - 1), i1x = min(i0x + 1, WW - 1);
  const float fy = (my & 1) ? 0.5f : 0.0f;
  const float fx = (mx & 1) ? 0.5f : 0.0f;

  const float* L = logits + (size_t)ni * HW;
  const float v00 = L[i0y * WW + i0x];
  const float v01 = L[i0y * WW + i1x];
  const float v10 = L[i1y * WW + i0x];
  const float v11 = L[i1y * WW + i1x];
  const float top = v00 + fx * (v01 - v00);
  const float bot = v10 + fx * (v11 - v10);
  out[idx] = top + fy * (bot - top);
}

extern "C" void kernel_launch(void* const* d_in, const int* in_sizes, int n_in,
                              void* d_out, int out_size, void* d_ws, size_t ws_size,
                              hipStream_t stream) {
  const float* mask_feats  = (const float*)d_in[0]; // (N,8,128,128) f32
  const float* mask_shift  = (const float*)d_in[1]; // (16384,2) f32
  const float* shifts      = (const float*)d_in[2]; // (n,2) f32
  const float* inst_params = (const float*)d_in[3]; // (n,169) f32
  const int*   im_inds     = (const int*)d_in[4];   // (n,) i32
  const int*   fpn_levels  = (const int*)d_in[5];   // (n,) i32
  // d_in[6] = mask_feat_stride == 8 -> upsample factor 2 (fixed by harness)

  const int n_inst = in_sizes[3] / NPARAMS;
  float* logits = (float*)d_ws;                     // n_inst * 16384 floats

  dim3 gridA(n_inst, 8);
  mask_head_logits<<<gridA, 256, 0, stream>>>(mask_feats, mask_shift, shifts,
                                              inst_params, im_inds, fpn_levels,
                                              logits);

  const int total = n_inst * 256 * 256;
  upsample2x<<<(total + 255) / 256, 256, 0, stream>>>(logits, (float*)d_out, total);
}